// MultiInstancePriorFilter_33380485824748
// MI455X (gfx1250) — compile-verified
//
#include <hip/hip_runtime.h>

// ---------------------------------------------------------------------------
// MultiInstancePriorFilter on MI455X (gfx1250, wave32)
//
// Pairwise containment sweep (8 x 4096 x 4096 pairs). Compute-bound in VALU;
// memory traffic (~1.3 MB total) is irrelevant at 23.3 TB/s. Strategy:
//   * stage j-tiles (boxes + class ids) into LDS via CDNA5 async-to-LDS DMA
//   * f32 VALU containment compares (exact, matches reference semantics)
//   * row-sum reduction over j via V_WMMA_F32_16X16X4_F32 with an all-ones
//     B matrix: D[m,n] += sum_k A[m,k] -- bit-exact f32 accumulate, and the
//     16x16 f32 accumulator doubles as 16 per-row running sums in VGPRs.
// ---------------------------------------------------------------------------

typedef __attribute__((ext_vector_type(2))) float v2f;
typedef __attribute__((ext_vector_type(8))) float v8f;
typedef __attribute__((ext_vector_type(4))) int   v4i;

#define AS1 __attribute__((address_space(1)))
#define AS3 __attribute__((address_space(3)))

static constexpr int   kB            = 8;
static constexpr int   kN            = 4096;
static constexpr int   kJTile        = 2048;   // j-boxes staged per LDS pass
static constexpr int   kThreads      = 256;    // 8 wave32 waves
static constexpr int   kRowsPerBlock = 128;    // 8 waves * 16 i-rows each
static constexpr float kThresh       = 0.8f;
static constexpr float kEps          = 1e-9f;

#if __has_builtin(__builtin_amdgcn_global_load_async_to_lds_b128) && \
    __has_builtin(__builtin_amdgcn_global_load_async_to_lds_b32)
#define HAVE_ASYNC_LDS 1
#else
#define HAVE_ASYNC_LDS 0
#endif

#if __has_builtin(__builtin_amdgcn_wmma_f32_16x16x4_f32)
#define HAVE_WMMA_F32X4 1
#else
#define HAVE_WMMA_F32X4 0
#endif

__device__ __forceinline__ void wait_async_copies() {
#if HAVE_ASYNC_LDS
#if __has_builtin(__builtin_amdgcn_s_wait_asynccnt)
  __builtin_amdgcn_s_wait_asynccnt(0);
#else
  asm volatile("s_wait_asynccnt 0" ::: "memory");
#endif
#endif
}

__global__ __launch_bounds__(kThreads) void mipf_kernel(
    const float* __restrict__ boxes,   // [B,N,4] x1,y1,x2,y2
    const int*   __restrict__ cls,     // [B,N]
    float* __restrict__ outBoxes,      // [B,N,4]
    float* __restrict__ outKeep)       // [B,N]
{
  __shared__ float4 sBox[kJTile];
  __shared__ int    sCls[kJTile];
  __shared__ float  sSum[kRowsPerBlock];

  const int b    = blockIdx.y;
  const int tid  = threadIdx.x;
  const int wave = tid >> 5;      // 0..7
  const int lane = tid & 31;      // wave32 lane
  const int row  = lane & 15;     // i-row within the wave's 16x16 WMMA tile
  const int half = lane >> 4;     // 0: K-slots {0,1}, 1: K-slots {2,3}
  const int iG   = blockIdx.x * kRowsPerBlock + wave * 16 + row;

  const float4* gBox = reinterpret_cast<const float4*>(boxes) + (size_t)b * kN;
  const int*    gCls = cls + (size_t)b * kN;

  // This lane's i-box (lanes L and L+16 both own row L).
  const float4 bi    = gBox[iG];
  const int    ci    = gCls[iG];
  const float  areaI = (bi.z - bi.x) * (bi.w - bi.y);

#if HAVE_WMMA_F32X4
  v8f acc0 = {0.f, 0.f, 0.f, 0.f, 0.f, 0.f, 0.f, 0.f};
  v8f acc1 = {0.f, 0.f, 0.f, 0.f, 0.f, 0.f, 0.f, 0.f};
  const v2f onesB = {1.0f, 1.0f};  // B = 4x16 all-ones -> D accumulates row sums
#else
  float accS = 0.0f;
#endif

  for (int jt = 0; jt < kN; jt += kJTile) {
    // ---- stage the j-tile into LDS (async DMA on CDNA5) -------------------
#pragma unroll
    for (int t = 0; t < kJTile / kThreads; ++t) {
      const int idx = tid + t * kThreads;
#if HAVE_ASYNC_LDS
      __builtin_amdgcn_global_load_async_to_lds_b128(
          (AS1 v4i*)(gBox + jt + idx), (AS3 v4i*)(&sBox[idx]), 0, 0);
      __builtin_amdgcn_global_load_async_to_lds_b32(
          (AS1 int*)(gCls + jt + idx), (AS3 int*)(&sCls[idx]), 0, 0);
#else
      sBox[idx] = gBox[jt + idx];
      sCls[idx] = gCls[jt + idx];
#endif
    }
    wait_async_copies();
    __syncthreads();

    // Overlap: speculatively prefetch next tile into cache during compute.
    if (jt + kJTile < kN) {
      __builtin_prefetch(gBox + jt + kJTile + tid * 4, 0, 1);
    }

    // ---- containment sweep over this tile ---------------------------------
    // Each iteration: two K=4 chunks (two independent WMMA accumulators).
    auto makeA = [&](int kbase) -> v2f {
      const int    k0 = kbase + (half << 1);  // lane's first K slot (A layout)
      const float4 b0 = sBox[k0];
      const float4 b1 = sBox[k0 + 1];
      const int    c0 = sCls[k0];
      const int    c1 = sCls[k0 + 1];
      const int    j0 = jt + k0;
      const bool v0 = (b0.x >= bi.x) & (b0.y >= bi.y) &
                      (b0.z <= bi.z) & (b0.w <= bi.w) &
                      (c0 == ci) & (j0 != iG);
      const bool v1 = (b1.x >= bi.x) & (b1.y >= bi.y) &
                      (b1.z <= bi.z) & (b1.w <= bi.w) &
                      (c1 == ci) & ((j0 + 1) != iG);
      v2f a;
      a.x = v0 ? (b0.z - b0.x) * (b0.w - b0.y) : 0.0f;
      a.y = v1 ? (b1.z - b1.x) * (b1.w - b1.y) : 0.0f;
      return a;
    };

#pragma unroll 2
    for (int jj = 0; jj < kJTile; jj += 8) {
      const v2f aA = makeA(jj);
      const v2f aB = makeA(jj + 4);
#if HAVE_WMMA_F32X4
      // D = A x ones + C : exact f32 row-sum accumulate (16 rows x 4 K).
      acc0 = __builtin_amdgcn_wmma_f32_16x16x4_f32(
          false, aA, false, onesB, (short)0, acc0, false, false);
      acc1 = __builtin_amdgcn_wmma_f32_16x16x4_f32(
          false, aB, false, onesB, (short)0, acc1, false, false);
#else
      accS += aA.x + aA.y + aB.x + aB.y;
#endif
    }
    __syncthreads();  // protect LDS before next tile's staging
  }

  // ---- epilogue: extract per-row sums, apply threshold, write outputs -----
#if HAVE_WMMA_F32X4
  v8f accT = acc0 + acc1;
  // C/D layout: lane 0 column holds rows 0..7 in VGPRs 0..7; lane 16 holds
  // rows 8..15. Every column is identical (B was all-ones), so column 0 works.
  if (row == 0) {  // lanes 0 and 16 of each wave
    const int base = wave * 16 + half * 8;
#pragma unroll
    for (int m = 0; m < 8; ++m) sSum[base + m] = accT[m];
  }
  __syncthreads();
  const float sumC = sSum[wave * 16 + row];
#else
  // Lanes L and L+16 hold complementary K-slot partial sums for row L.
  const float sumC = accS + __shfl_xor(accS, 16, 32);
#endif

  if (half == 0) {  // one writer per i-row
    const float keep = (sumC <= kThresh * (areaI + kEps)) ? 1.0f : 0.0f;
    float4 ob;
    ob.x = bi.x * keep; ob.y = bi.y * keep;
    ob.z = bi.z * keep; ob.w = bi.w * keep;
    reinterpret_cast<float4*>(outBoxes)[(size_t)b * kN + iG] = ob;
    outKeep[(size_t)b * kN + iG] = keep;
  }
}

extern "C" void kernel_launch(void* const* d_in, const int* in_sizes, int n_in,
                              void* d_out, int out_size, void* d_ws, size_t ws_size,
                              hipStream_t stream) {
  (void)in_sizes; (void)n_in; (void)out_size; (void)d_ws; (void)ws_size;
  const float* boxes = (const float*)d_in[0];
  const int*   cls   = (const int*)d_in[1];
  float* outBoxes = (float*)d_out;                              // B*N*4 floats
  float* outKeep  = (float*)d_out + (size_t)kB * kN * 4;        // then B*N keep

  dim3 grid(kN / kRowsPerBlock, kB);
  mipf_kernel<<<grid, dim3(kThreads), 0, stream>>>(boxes, cls, outBoxes, outKeep);
}